// AutoCorrelation_85478439125334
// MI455X (gfx1250) — compile-verified
//
#include <hip/hip_runtime.h>
#include <hip/hip_bf16.h>

#define B_  8
#define H_  8
#define L_  2048
#define D_  64
#define C_  512      // H_*D_  (flattened head*channel inner dim)
#define NK_ 16       // FACTOR * ceil(log(2048)) = 2*8 = 16 delays

typedef __attribute__((ext_vector_type(2))) float v2f;
typedef __attribute__((ext_vector_type(8))) float v8f;

#define WMMA_F32(A, Bv, Cv) \
    __builtin_amdgcn_wmma_f32_16x16x4_f32(false, (A), false, (Bv), (short)0, (Cv), false, false)

// ---------------------------------------------------------------------------
// Stage 0: zero the mean-correlation accumulator in workspace
// ---------------------------------------------------------------------------
__global__ void ac_zero_kernel(float* __restrict__ buf, int n) {
    int i = blockIdx.x * blockDim.x + threadIdx.x;
    if (i < n) buf[i] = 0.0f;
}

// ---------------------------------------------------------------------------
// Stage 1: WMMA correlation with 2x2 register blocking.
// One wave (32 threads) computes a 32x32 region of S[b] = Q[b]*K[b]^T
// (inner dim C_=512, fp32 via V_WMMA_F32_16X16X4_F32) as four 16x16 tiles,
// reusing each A/B fragment twice (1.0 loads per WMMA).  The 63 diagonal
// bands of the region are reduced in LDS (ds_add_f32), then accumulated
// into meanCorr[b][tau] with tau = (n - m) mod L via global fp32 atomics.
// ---------------------------------------------------------------------------
__global__ void __launch_bounds__(32)
ac_corr_wmma_kernel(const float* __restrict__ q,
                    const float* __restrict__ k,
                    float* __restrict__ meanCorr) {
    const int tm   = blockIdx.x;     // 32-row block index (query n)
    const int tn   = blockIdx.y;     // 32-col block index (key m)
    const int b    = blockIdx.z;     // batch
    const int lane = threadIdx.x;    // 0..31 (wave32)
    const int rc   = lane & 15;      // row (A) / col (B) within a 16-tile
    const int half = lane >> 4;      // selects K-pair {0,1} vs {2,3}

    const size_t bofs = (size_t)b * H_ * L_ * D_;
    const int n0 = tm * 32 + rc;     // A fragment rows (tile row 0 / 1)
    const int m0 = tn * 32 + rc;     // B fragment cols (tile col 0 / 1)

    // loop-invariant per-lane base pointers (lane's K slot = +2*half)
    const float* qa0 = q + bofs + (size_t)n0 * D_ + 2 * half;
    const float* qa1 = qa0 + (size_t)16 * D_;
    const float* kb0 = k + bofs + (size_t)m0 * D_ + 2 * half;
    const float* kb1 = kb0 + (size_t)16 * D_;

    v8f acc00 = {}, acc01 = {}, acc10 = {}, acc11 = {};

    for (int h = 0; h < H_; ++h) {
        const size_t hofs = (size_t)h * L_ * D_;
        const float* pq0 = qa0 + hofs;
        const float* pq1 = qa1 + hofs;
        const float* pk0 = kb0 + hofs;
        const float* pk1 = kb1 + hofs;
#pragma unroll
        for (int dd = 0; dd < D_; dd += 4) {        // 16 K-steps, imm offsets
            const v2f a0 = *reinterpret_cast<const v2f*>(pq0 + dd);
            const v2f a1 = *reinterpret_cast<const v2f*>(pq1 + dd);
            const v2f b0 = *reinterpret_cast<const v2f*>(pk0 + dd);
            const v2f b1 = *reinterpret_cast<const v2f*>(pk1 + dd);
            acc00 = WMMA_F32(a0, b0, acc00);
            acc01 = WMMA_F32(a0, b1, acc01);
            acc10 = WMMA_F32(a1, b0, acc10);
            acc11 = WMMA_F32(a1, b1, acc11);
        }
    }

    // ---- 63-band diagonal reduction of the 32x32 region, in LDS ----
    __shared__ float bandsum[64];
    bandsum[lane]      = 0.0f;
    bandsum[lane + 32] = 0.0f;
    __syncthreads();

    // C/D layout: element v of lane holds local row (16*ti + v + 8*half),
    //             local col (16*tj + rc);  band li = nl - ml + 31 in [0,62].
#define AC_REDUCE(ACC, TI, TJ)                                              \
    {                                                                       \
        _Pragma("unroll")                                                   \
        for (int v = 0; v < 8; ++v) {                                       \
            const int li = (16 * (TI) + v + 8 * half) - (16 * (TJ) + rc) + 31; \
            atomicAdd(&bandsum[li], (ACC)[v]);                              \
        }                                                                   \
    }
    AC_REDUCE(acc00, 0, 0)
    AC_REDUCE(acc01, 0, 1)
    AC_REDUCE(acc10, 1, 0)
    AC_REDUCE(acc11, 1, 1)
#undef AC_REDUCE
    __syncthreads();

#pragma unroll
    for (int p = 0; p < 2; ++p) {
        const int li = lane + 32 * p;
        if (li < 63) {
            const int tau = (32 * (tm - tn) + li - 31) & (L_ - 1);
            atomicAdd(&meanCorr[b * L_ + tau], bandsum[li] * (1.0f / (float)C_));
        }
    }
}

// ---------------------------------------------------------------------------
// Stage 2: per-batch top-16 of meanCorr[b][0..2047] + softmax of the scores.
// One 256-thread workgroup per batch.
// ---------------------------------------------------------------------------
__global__ void ac_topk_softmax_kernel(const float* __restrict__ meanCorr,
                                       float* __restrict__ w,
                                       int* __restrict__ delays) {
    const int b = blockIdx.x;
    const int t = threadIdx.x;                  // 0..255

    __shared__ float vals[L_];
    __shared__ float rmax[256];
    __shared__ int   ridx[256];
    __shared__ float topv[NK_];
    __shared__ int   topi[NK_];

    for (int i = t; i < L_; i += 256) vals[i] = meanCorr[b * L_ + i];
    __syncthreads();

    for (int kk = 0; kk < NK_; ++kk) {
        float best = -3.4e38f;
        int   bi   = 0;
        for (int i = t; i < L_; i += 256) {
            float x = vals[i];
            if (x > best) { best = x; bi = i; }
        }
        rmax[t] = best; ridx[t] = bi;
        __syncthreads();
        for (int s = 128; s > 0; s >>= 1) {
            if (t < s && rmax[t + s] > rmax[t]) {
                rmax[t] = rmax[t + s];
                ridx[t] = ridx[t + s];
            }
            __syncthreads();
        }
        if (t == 0) {
            topv[kk] = rmax[0];
            topi[kk] = ridx[0];
            vals[ridx[0]] = -3.4e38f;           // remove winner
        }
        __syncthreads();
    }

    if (t == 0) {
        float mx = topv[0];                     // max is first (descending)
        float e[NK_], s = 0.0f;
#pragma unroll
        for (int kk = 0; kk < NK_; ++kk) { e[kk] = __expf(topv[kk] - mx); s += e[kk]; }
        const float inv = 1.0f / s;
#pragma unroll
        for (int kk = 0; kk < NK_; ++kk) {
            w[b * NK_ + kk]      = e[kk] * inv;
            delays[b * NK_ + kk] = topi[kk];
        }
    }
}

// ---------------------------------------------------------------------------
// Stage 3: context[b,h,l,:] = sum_k w[b,k] * value[b,h,(l+delay_k)%L,:]
// One thread per 4 consecutive d-channels (float4). Bandwidth-bound.
// ---------------------------------------------------------------------------
__global__ void ac_combine_kernel(const float* __restrict__ value,
                                  const float* __restrict__ w,
                                  const int* __restrict__ delays,
                                  float* __restrict__ out) {
    const size_t tid = (size_t)blockIdx.x * blockDim.x + threadIdx.x;
    // tid = ((bh*L_) + l)*16 + d4,  bh in [0,64)
    const int d4 = (int)(tid & 15);
    const int l  = (int)((tid >> 4) & (L_ - 1));
    const int bh = (int)(tid >> 15);
    const int b  = bh >> 3;

    const float* wb  = w + b * NK_;
    const int*   db  = delays + b * NK_;
    const float* row = value + (size_t)bh * L_ * D_ + d4 * 4;

    float4 acc = make_float4(0.f, 0.f, 0.f, 0.f);
#pragma unroll
    for (int kk = 0; kk < NK_; ++kk) {
        const int src = (l + db[kk]) & (L_ - 1);
        const float4 vv = *reinterpret_cast<const float4*>(row + (size_t)src * D_);
        const float wk = wb[kk];
        acc.x += wk * vv.x; acc.y += wk * vv.y;
        acc.z += wk * vv.z; acc.w += wk * vv.w;
    }
    *reinterpret_cast<float4*>(out + tid * 4) = acc;
}

// ---------------------------------------------------------------------------
extern "C" void kernel_launch(void* const* d_in, const int* in_sizes, int n_in,
                              void* d_out, int out_size, void* d_ws, size_t ws_size,
                              hipStream_t stream) {
    (void)in_sizes; (void)n_in; (void)out_size; (void)ws_size;
    const float* q = (const float*)d_in[0];
    const float* k = (const float*)d_in[1];
    const float* v = (const float*)d_in[2];
    float* out = (float*)d_out;

    // workspace layout: meanCorr (B*L floats) | w (B*16 floats) | delays (B*16 ints)
    float* meanCorr = (float*)d_ws;
    float* w        = meanCorr + B_ * L_;
    int*   delays   = (int*)(w + B_ * NK_);

    // stage 0: zero accumulator
    ac_zero_kernel<<<(B_ * L_ + 255) / 256, 256, 0, stream>>>(meanCorr, B_ * L_);

    // stage 1: WMMA correlation (one wave per 32x32 block, 2x2 tile blocking)
    dim3 g1(L_ / 32, L_ / 32, B_);                 // 64 x 64 x 8
    ac_corr_wmma_kernel<<<g1, 32, 0, stream>>>(q, k, meanCorr);

    // stage 2: top-16 + softmax, one workgroup per batch
    ac_topk_softmax_kernel<<<B_, 256, 0, stream>>>(meanCorr, w, delays);

    // stage 3: weighted circular-shift combine
    const size_t total = (size_t)B_ * H_ * L_ * (D_ / 4);   // 2,097,152 threads
    ac_combine_kernel<<<(unsigned)(total / 256), 256, 0, stream>>>(v, w, delays, out);
}